// PathFinder_22067541967022
// MI455X (gfx1250) — compile-verified
//
#include <hip/hip_runtime.h>
#include <math.h>

#define N_NODES 100000
#define N_EDGES 3200000
#define D 768
#define FF 1536
#define NPATH 5
#define LPATH 4
#define MP 32           // padded row count (20 real rows -> 2 M-tiles of 16)
#define KCHUNK 64

typedef __attribute__((ext_vector_type(2))) float v2f;
typedef __attribute__((ext_vector_type(4))) float f4;
typedef __attribute__((ext_vector_type(8))) float v8f;
typedef __attribute__((ext_vector_type(4))) unsigned int u32x4;
typedef __attribute__((ext_vector_type(8))) int i32x8;
typedef __attribute__((ext_vector_type(4))) int i32x4;

// ---------------- small utility kernels ----------------

__global__ void k_zero_i(int* p, int n) {
  for (int i = blockIdx.x * blockDim.x + threadIdx.x; i < n; i += gridDim.x * blockDim.x)
    p[i] = 0;
}

// qp = Wq @ query + bq   (768 rows, wave per row)
__global__ void k_qproj(const float* __restrict__ Wq, const float* __restrict__ q,
                        const float* __restrict__ bq, float* __restrict__ qp) {
  int tid = threadIdx.x, wave = tid >> 5, lane = tid & 31;
  int row = blockIdx.x * 8 + wave;
  float s = 0.f;
  #pragma unroll
  for (int i = 0; i < 24; ++i) { int k = lane + i * 32; s += Wq[row * D + k] * q[k]; }
  for (int off = 16; off; off >>= 1) s += __shfl_xor(s, off, 32);
  if (lane == 0) qp[row] = s + bq[row];
}

// sims[i] = dot(node_features[i], qp)  -- bandwidth-bound GEMV, NT loads
__global__ void k_sims(const float* __restrict__ nf, const float* __restrict__ qp,
                       float* __restrict__ sims) {
  __shared__ float q[D];
  int tid = threadIdx.x;
  for (int k = tid; k < D; k += 256) q[k] = qp[k];
  __syncthreads();
  int wave = tid >> 5, lane = tid & 31;
  long long row = (long long)blockIdx.x * 8 + wave;
  if (row >= N_NODES) return;
  const f4* base = (const f4*)(nf + row * D);
  float s = 0.f;
  #pragma unroll
  for (int i = 0; i < 6; ++i) {
    f4 v = __builtin_nontemporal_load(&base[lane + i * 32]);
    int k = (lane + i * 32) * 4;
    s += v.x * q[k] + v.y * q[k + 1] + v.z * q[k + 2] + v.w * q[k + 3];
  }
  for (int off = 16; off; off >>= 1) s += __shfl_xor(s, off, 32);
  if (lane == 0) sims[row] = s;
}

// sequential 5-pass argmax top-k (descending, lower index wins ties) -> meta[0..4]
__global__ void k_topk(const float* __restrict__ sims, int* __restrict__ meta) {
  __shared__ float bv[1024];
  __shared__ int bi[1024];
  __shared__ int sel[NPATH];
  int tid = threadIdx.x;
  for (int t = 0; t < NPATH; ++t) {
    float best = -1e30f; int bidx = 0x7fffffff;
    for (int i = tid; i < N_NODES; i += 1024) {
      bool skip = false;
      for (int s = 0; s < t; ++s) if (sel[s] == i) skip = true;
      if (skip) continue;
      float v = sims[i];
      if (v > best || (v == best && i < bidx)) { best = v; bidx = i; }
    }
    bv[tid] = best; bi[tid] = bidx;
    __syncthreads();
    for (int off = 512; off; off >>= 1) {
      if (tid < off) {
        float ov = bv[tid + off]; int oi = bi[tid + off];
        if (ov > bv[tid] || (ov == bv[tid] && oi < bi[tid])) { bv[tid] = ov; bi[tid] = oi; }
      }
      __syncthreads();
    }
    if (tid == 0) { sel[t] = bi[0]; meta[t] = bi[0]; }
    __syncthreads();
  }
}

// ---------------- CSR build (edge-order adjacency) ----------------

__global__ void k_deg(const int* __restrict__ src, int* __restrict__ deg) {
  for (int e = blockIdx.x * blockDim.x + threadIdx.x; e < N_EDGES; e += gridDim.x * blockDim.x)
    atomicAdd(&deg[src[e]], 1);
}

__global__ void k_scan(const int* __restrict__ deg, int* __restrict__ offs) {
  __shared__ int part[257];
  int tid = threadIdx.x;
  const int chunk = (N_NODES + 255) / 256;
  int beg = tid * chunk, s = 0;
  for (int i = 0; i < chunk; ++i) { int idx = beg + i; if (idx < N_NODES) s += deg[idx]; }
  part[tid] = s;
  __syncthreads();
  if (tid == 0) {
    int run = 0;
    for (int i = 0; i < 256; ++i) { int t = part[i]; part[i] = run; run += t; }
    part[256] = run;
  }
  __syncthreads();
  int run = part[tid];
  for (int i = 0; i < chunk; ++i) {
    int idx = beg + i;
    if (idx < N_NODES) { offs[idx] = run; run += deg[idx]; }
  }
  if (tid == 255) offs[N_NODES] = part[256];
}

__global__ void k_scatter(const int* __restrict__ src, const int* __restrict__ dst,
                          const int* __restrict__ offs, int* __restrict__ cursor,
                          int* __restrict__ eid, int* __restrict__ cdst) {
  for (int e = blockIdx.x * blockDim.x + threadIdx.x; e < N_EDGES; e += gridDim.x * blockDim.x) {
    int s = src[e];
    int pos = offs[s] + atomicAdd(&cursor[s], 1);
    eid[pos] = e;
    cdst[pos] = dst[e];
  }
}

// restore edge order per node (stable-sort-by-src equivalence)
__global__ void k_sortcsr(const int* __restrict__ offs, int* __restrict__ eid,
                          int* __restrict__ cdst) {
  int v = blockIdx.x * blockDim.x + threadIdx.x;
  if (v >= N_NODES) return;
  int lo = offs[v], hi = offs[v + 1];
  for (int i = lo + 1; i < hi; ++i) {
    int e = eid[i], d = cdst[i], j = i - 1;
    while (j >= lo && eid[j] > e) { eid[j + 1] = eid[j]; cdst[j + 1] = cdst[j]; --j; }
    eid[j + 1] = e; cdst[j + 1] = d;
  }
}

// greedy path expansion (exact reference semantics), single thread
__global__ void k_expand(const int* __restrict__ offs, const int* __restrict__ cdst,
                         int* __restrict__ meta) {
  if (threadIdx.x != 0 || blockIdx.x != 0) return;
  int* pathidx = meta + 8;
  int ng = 0;
  int cur[NPATH][LPATH], nxt[NPATH][LPATH];
  for (int s = 0; s < NPATH && ng < NPATH; ++s) {
    cur[0][0] = meta[s];
    int ncur = 1, hop;
    for (hop = 0; hop < 3 && ncur > 0; ++hop) {
      int nn = 0;
      for (int i = 0; i < ncur && nn < NPATH; ++i) {
        int v = cur[i][hop];
        int lo = offs[v], hi = offs[v + 1];
        for (int t = lo; t < hi && nn < NPATH; ++t) {
          int nb = cdst[t];
          bool inp = false;
          for (int q = 0; q <= hop; ++q) if (cur[i][q] == nb) { inp = true; break; }
          if (!inp) {
            for (int q = 0; q <= hop; ++q) nxt[nn][q] = cur[i][q];
            nxt[nn][hop + 1] = nb;
            ++nn;
          }
        }
      }
      for (int i = 0; i < nn; ++i)
        for (int q = 0; q <= hop + 1; ++q) cur[i][q] = nxt[i][q];
      ncur = nn;
    }
    if (ncur > 0 && hop == 3) {
      for (int i = 0; i < ncur && ng < NPATH; ++i) {
        for (int q = 0; q < LPATH; ++q) pathidx[ng * LPATH + q] = cur[i][q];
        ++ng;
      }
    }
  }
  meta[30] = ng;
}

// x[row][k] = node_features[path[row]][k] + posenc(l,k); zero pad rows
__global__ void k_gather(const float* __restrict__ nf, const int* __restrict__ meta,
                         float* __restrict__ x) {
  int row = blockIdx.x;                 // 0..31
  int p = row >> 2, l = row & 3;
  int nP = meta[30];
  for (int k = threadIdx.x; k < D; k += blockDim.x) {
    float v = 0.f;
    if (row < NPATH * LPATH && p < nP) {
      int node = meta[8 + p * LPATH + l];
      float i2 = (float)((k >> 1) * 2);
      float ang = (float)l * expf(-logf(10000.f) * i2 / (float)D);
      float pe = (k & 1) ? cosf(ang) : sinf(ang);
      v = nf[(long long)node * D + k] + pe;
    }
    x[row * D + k] = v;
  }
}

// ---------------- WMMA GEMM with TDM weight staging ----------------
// Y[32,N] = act(X[32,K] @ W[N,K]^T + bias)
// block = 64 threads (2 waves: M-tiles 0..15 and 16..31), grid.x = N/16.
// Weight tiles (16 x 64 fp32) are DMA'd into LDS by the Tensor Data Mover,
// double-buffered; D#.pad (4 DW every 64 DW) keeps the LDS pitch at 68 floats
// so compute reads stay bank-conflict free. Synchronized via s_wait_tensorcnt.
__device__ __forceinline__ void tdm_fetch_w(const float* W, long long elem_off, int K,
                                            unsigned lds_off) {
  unsigned long long gaddr =
      (unsigned long long)(size_t)W + (unsigned long long)elem_off * 4ull;
  u32x4 g0;
  g0.x = 1u;                                            // count=1, load descriptor
  g0.y = lds_off;                                       // LDS byte offset (flat low32)
  g0.z = (unsigned)gaddr;                               // global_addr[31:0]
  g0.w = (unsigned)((gaddr >> 32) & 0x01FFFFFFull) | 0x80000000u;  // [56:32] | type=2
  i32x8 g1;
  g1[0] = (2 << 16)       // data_size = 4 bytes
        | (1 << 20)       // pad_enable
        | (5 << 22)       // pad_interval: 64 DWORDs
        | (3 << 25);      // pad_amount: 4 DWORDs
  g1[1] = (KCHUNK & 0xFFFF) << 16;                      // tensor_dim0 = 64
  g1[2] = (KCHUNK >> 16) | (16 << 16);                  // tensor_dim1 = 16
  g1[3] = (KCHUNK << 16);                               // tile_dim0 = 64
  g1[4] = 16;                                           // tile_dim1 = 16, tile_dim2 = 0
  g1[5] = K;                                            // tensor_dim0_stride lo
  g1[6] = 0;                                            // stride hi / dim1_stride lo
  g1[7] = 0;
  i32x4 z4 = {0, 0, 0, 0};                              // 2-D tensor: groups 2/3 unused
  i32x8 z8 = {0, 0, 0, 0, 0, 0, 0, 0};
  __builtin_amdgcn_tensor_load_to_lds(g0, g1, z4, z4, z8, 0);
}

__global__ void k_gemm(const float* __restrict__ X, const float* __restrict__ W,
                       const float* __restrict__ bias, float* __restrict__ Y,
                       int K, int N, int relu) {
  __shared__ float Xs[MP][KCHUNK + 4];
  __shared__ float Ws[2][16][KCHUNK + 4];
  const int n0 = blockIdx.x * 16;
  const int tid = threadIdx.x;
  const int wave = tid >> 5;
  const int lane = tid & 31;
  const int half = lane >> 4;
  const int lr = lane & 15;
  const int m0 = wave * 16;
  const int nchunk = K / KCHUNK;

  // flat LDS pointers carry the LDS byte offset in addr[31:0] (ISA 10.2)
  const unsigned ws_off0 = (unsigned)(size_t)&Ws[0][0][0];
  const unsigned ws_off1 = (unsigned)(size_t)&Ws[1][0][0];

  if (wave == 0)  // prefetch weight chunk 0 (TDM ignores EXEC; uniform branch skips wave1)
    tdm_fetch_w(W, (long long)n0 * K, K, ws_off0);

  v8f acc = {0.f, 0.f, 0.f, 0.f, 0.f, 0.f, 0.f, 0.f};

  for (int c = 0; c < nchunk; ++c) {
    const int kc = c * KCHUNK;
    if (wave == 0 && c + 1 < nchunk)
      tdm_fetch_w(W, (long long)n0 * K + kc + KCHUNK, K, ((c + 1) & 1) ? ws_off1 : ws_off0);
    for (int r = 0; r < MP; ++r) Xs[r][tid] = X[r * K + kc + tid];
    if (wave == 0) {
      if (c + 1 < nchunk) __builtin_amdgcn_s_wait_tensorcnt(1);  // oldest TDM done
      else                __builtin_amdgcn_s_wait_tensorcnt(0);
    }
    __syncthreads();
    const float (*Wb)[KCHUNK + 4] = Ws[c & 1];
    #pragma unroll
    for (int k0 = 0; k0 < KCHUNK; k0 += 4) {
      v2f a, b;
      // A 16x4 f32 layout: VGPR0 = K{k0, k0+2} (lo/hi half-wave), VGPR1 = K{k0+1, k0+3}
      a.x = Xs[m0 + lr][k0 + 2 * half];
      a.y = Xs[m0 + lr][k0 + 2 * half + 1];
      // B 4x16: B[k][n] = W[n][k]
      b.x = Wb[lr][k0 + 2 * half];
      b.y = Wb[lr][k0 + 2 * half + 1];
      acc = __builtin_amdgcn_wmma_f32_16x16x4_f32(false, a, false, b, (short)0, acc,
                                                  false, false);
    }
    __syncthreads();
  }
  const int col = n0 + lr;
  const float bv = bias ? bias[col] : 0.f;
  #pragma unroll
  for (int r = 0; r < 8; ++r) {
    int row = m0 + half * 8 + r;       // C VGPR r: lanes0-15 -> M=r, lanes16-31 -> M=r+8
    float v = acc[r] + bv;
    if (relu) v = fmaxf(v, 0.f);
    Y[row * N + col] = v;
  }
}

// ---------------- attention: 4x4 per (path, head) ----------------
__global__ void k_attn(const float* __restrict__ qkv, float* __restrict__ out) {
  int p = blockIdx.x >> 3, h = blockIdx.x & 7;
  int j = threadIdx.x;                   // 0..95
  __shared__ float Q[4][96], Kk[4][96], V[4][96];
  __shared__ float S[4][4];
  for (int l = 0; l < 4; ++l) {
    int row = p * 4 + l;
    Q[l][j]  = qkv[row * 2304 + h * 96 + j];
    Kk[l][j] = qkv[row * 2304 + D + h * 96 + j];
    V[l][j]  = qkv[row * 2304 + 2 * D + h * 96 + j];
  }
  __syncthreads();
  if (j < 16) {
    int l = j >> 2, m = j & 3;
    float s = 0.f;
    for (int d = 0; d < 96; ++d) s += Q[l][d] * Kk[m][d];
    S[l][m] = s * 0.10206207261596575f;  // 1/sqrt(96)
  }
  __syncthreads();
  if (j < 4) {
    float mx = fmaxf(fmaxf(S[j][0], S[j][1]), fmaxf(S[j][2], S[j][3]));
    float e0 = expf(S[j][0] - mx), e1 = expf(S[j][1] - mx);
    float e2 = expf(S[j][2] - mx), e3 = expf(S[j][3] - mx);
    float inv = 1.f / (e0 + e1 + e2 + e3);
    S[j][0] = e0 * inv; S[j][1] = e1 * inv; S[j][2] = e2 * inv; S[j][3] = e3 * inv;
  }
  __syncthreads();
  for (int l = 0; l < 4; ++l) {
    float o = 0.f;
    for (int m = 0; m < 4; ++m) o += S[l][m] * V[m][j];
    out[(p * 4 + l) * D + h * 96 + j] = o;
  }
}

// y = LN(a + r) * w + b   (block per row, 256 threads, 3 elems each)
__global__ void k_ln(const float* __restrict__ a, const float* __restrict__ r,
                     const float* __restrict__ w, const float* __restrict__ b,
                     float* __restrict__ y) {
  __shared__ float red[256];
  int row = blockIdx.x, tid = threadIdx.x;
  float xv[3], s = 0.f;
  #pragma unroll
  for (int i = 0; i < 3; ++i) {
    int k = tid + i * 256;
    xv[i] = a[row * D + k] + r[row * D + k];
    s += xv[i];
  }
  red[tid] = s; __syncthreads();
  for (int off = 128; off; off >>= 1) { if (tid < off) red[tid] += red[tid + off]; __syncthreads(); }
  float mean = red[0] / (float)D;
  __syncthreads();
  s = 0.f;
  #pragma unroll
  for (int i = 0; i < 3; ++i) { float d = xv[i] - mean; s += d * d; }
  red[tid] = s; __syncthreads();
  for (int off = 128; off; off >>= 1) { if (tid < off) red[tid] += red[tid + off]; __syncthreads(); }
  float rstd = rsqrtf(red[0] / (float)D + 1e-5f);
  #pragma unroll
  for (int i = 0; i < 3; ++i) {
    int k = tid + i * 256;
    y[row * D + k] = (xv[i] - mean) * rstd * w[k] + b[k];
  }
}

// repr[p] = mean over L of x[p*4+l]; pad rows zero
__global__ void k_mean(const float* __restrict__ x, float* __restrict__ repr) {
  int p = blockIdx.x;                    // 0..31
  for (int k = threadIdx.x; k < D; k += blockDim.x) {
    float v = 0.f;
    if (p < NPATH)
      v = 0.25f * (x[(4 * p) * D + k] + x[(4 * p + 1) * D + k] +
                   x[(4 * p + 2) * D + k] + x[(4 * p + 3) * D + k]);
    repr[p * D + k] = v;
  }
}

// scores[p] = dot(sch[p], sc_w2) + sc_b2 (wave per path)
__global__ void k_score(const float* __restrict__ sch, const float* __restrict__ w2,
                        const float* __restrict__ b2, float* __restrict__ scores) {
  int tid = threadIdx.x, wave = tid >> 5, lane = tid & 31;
  if (wave >= NPATH) return;
  float s = 0.f;
  #pragma unroll
  for (int i = 0; i < 12; ++i) { int k = lane + i * 32; s += sch[wave * 384 + k] * w2[k]; }
  for (int off = 16; off; off >>= 1) s += __shfl_xor(s, off, 32);
  if (lane == 0) scores[wave] = s + b2[0];
}

// out[k] = sum_p softmax(scores[0..nP))_p * repr[p][k]
__global__ void k_final(const float* __restrict__ repr, const float* __restrict__ scores,
                        const int* __restrict__ meta, float* __restrict__ out) {
  int k = threadIdx.x;                   // blockDim = 768
  int nP = meta[30];
  if (nP == 0) { out[k] = 0.f; return; }
  float mx = -1e30f;
  for (int p = 0; p < nP; ++p) mx = fmaxf(mx, scores[p]);
  float w[NPATH], sum = 0.f;
  for (int p = 0; p < nP; ++p) { w[p] = expf(scores[p] - mx); sum += w[p]; }
  float inv = 1.f / sum, o = 0.f;
  for (int p = 0; p < nP; ++p) o += (w[p] * inv) * repr[p * D + k];
  out[k] = o;
}

// ---------------- host orchestration ----------------

extern "C" void kernel_launch(void* const* d_in, const int* in_sizes, int n_in,
                              void* d_out, int out_size, void* d_ws, size_t ws_size,
                              hipStream_t stream) {
  (void)in_sizes; (void)n_in; (void)out_size; (void)ws_size;
  const float* nf    = (const float*)d_in[0];
  const float* query = (const float*)d_in[1];
  const int*   esrc  = (const int*)d_in[2];
  const int*   edst  = esrc + N_EDGES;
  const float* Wq    = (const float*)d_in[4];
  const float* bq    = (const float*)d_in[5];
  const float* qkv_w = (const float*)d_in[6];
  const float* qkv_b = (const float*)d_in[7];
  const float* out_w = (const float*)d_in[8];
  const float* out_b = (const float*)d_in[9];
  const float* ln1_w = (const float*)d_in[10];
  const float* ln1_b = (const float*)d_in[11];
  const float* ff1_w = (const float*)d_in[12];
  const float* ff1_b = (const float*)d_in[13];
  const float* ff2_w = (const float*)d_in[14];
  const float* ff2_b = (const float*)d_in[15];
  const float* ln2_w = (const float*)d_in[16];
  const float* ln2_b = (const float*)d_in[17];
  const float* sc_w1 = (const float*)d_in[18];
  const float* sc_b1 = (const float*)d_in[19];
  const float* sc_w2 = (const float*)d_in[20];
  const float* sc_b2 = (const float*)d_in[21];

  char* base = (char*)d_ws;
  size_t o = 0;
  auto take = [&](size_t bytes) { char* p = base + o; o = (o + bytes + 255) & ~(size_t)255; return p; };
  float* qp     = (float*)take(D * 4);
  float* sims   = (float*)take((size_t)N_NODES * 4);
  int*   deg    = (int*)take((size_t)N_NODES * 4);
  int*   offs   = (int*)take((size_t)(N_NODES + 1) * 4);
  int*   cursor = (int*)take((size_t)N_NODES * 4);
  int*   meta   = (int*)take(64 * 4);
  int*   eid    = (int*)take((size_t)N_EDGES * 4);
  int*   cdst   = (int*)take((size_t)N_EDGES * 4);
  float* x      = (float*)take(MP * D * 4);
  float* qkv    = (float*)take(MP * 3 * D * 4);
  float* attno  = (float*)take(MP * D * 4);
  float* proj   = (float*)take(MP * D * 4);
  float* x1     = (float*)take(MP * D * 4);
  float* hb     = (float*)take(MP * FF * 4);
  float* h2     = (float*)take(MP * D * 4);
  float* repr   = (float*)take(MP * D * 4);
  float* sch    = (float*)take(MP * 384 * 4);
  float* scores = (float*)take(8 * 4);

  // 1) query projection + similarity GEMV + top-5
  k_qproj<<<96, 256, 0, stream>>>(Wq, query, bq, qp);
  k_sims<<<N_NODES / 8, 256, 0, stream>>>(nf, qp, sims);
  k_topk<<<1, 1024, 0, stream>>>(sims, meta);

  // 2) edge-order CSR adjacency
  k_zero_i<<<400, 256, 0, stream>>>(deg, N_NODES);
  k_deg<<<2048, 256, 0, stream>>>(esrc, deg);
  k_scan<<<1, 256, 0, stream>>>(deg, offs);
  k_zero_i<<<400, 256, 0, stream>>>(cursor, N_NODES);
  k_scatter<<<2048, 256, 0, stream>>>(esrc, edst, offs, cursor, eid, cdst);
  k_sortcsr<<<(N_NODES + 255) / 256, 256, 0, stream>>>(offs, eid, cdst);

  // 3) greedy path expansion + gather with positional encoding
  k_expand<<<1, 32, 0, stream>>>(offs, cdst, meta);
  k_gather<<<MP, 256, 0, stream>>>(nf, meta, x);

  // 4) two transformer encoder layers (WMMA f32 16x16x4 + TDM weight staging)
  for (int l = 0; l < 2; ++l) {
    k_gemm<<<3 * D / 16, 64, 0, stream>>>(x, qkv_w + (size_t)l * 3 * D * D,
                                          qkv_b + l * 3 * D, qkv, D, 3 * D, 0);
    k_attn<<<NPATH * 8, 96, 0, stream>>>(qkv, attno);
    k_gemm<<<D / 16, 64, 0, stream>>>(attno, out_w + (size_t)l * D * D,
                                      out_b + l * D, proj, D, D, 0);
    k_ln<<<MP, 256, 0, stream>>>(x, proj, ln1_w + l * D, ln1_b + l * D, x1);
    k_gemm<<<FF / 16, 64, 0, stream>>>(x1, ff1_w + (size_t)l * FF * D,
                                       ff1_b + l * FF, hb, D, FF, 1);
    k_gemm<<<D / 16, 64, 0, stream>>>(hb, ff2_w + (size_t)l * D * FF,
                                      ff2_b + l * D, h2, FF, D, 0);
    k_ln<<<MP, 256, 0, stream>>>(x1, h2, ln2_w + l * D, ln2_b + l * D, x);
  }

  // 5) path pooling + scorer + softmax-weighted sum
  k_mean<<<MP, 256, 0, stream>>>(x, repr);
  k_gemm<<<384 / 16, 64, 0, stream>>>(repr, sc_w1, sc_b1, sch, D, 384, 1);
  k_score<<<1, 256, 0, stream>>>(sch, sc_w2, sc_b2, scores);
  k_final<<<1, D, 0, stream>>>(repr, scores, meta, (float*)d_out);
}